// WaveletFilter_38182259262134
// MI455X (gfx1250) — compile-verified
//
#include <hip/hip_runtime.h>
#include <math.h>

#define NN   32768   // total nodes
#define CC   128     // channels
#define EE   524288  // total edges
#define NPGc 512     // nodes per graph
#define NBg  64      // graphs
#define NG   32      // gaussians

typedef float v2f __attribute__((ext_vector_type(2)));
typedef float v8f __attribute__((ext_vector_type(8)));

#define USE_ASYNC_LDS 1

__device__ __forceinline__ v8f wmma_f32(v2f a, v2f b, v8f c) {
  // D = A(16x4) * B(4x16) + C, fp32 WMMA (wave32)
  return __builtin_amdgcn_wmma_f32_16x16x4_f32(
      /*neg_a=*/false, a, /*neg_b=*/false, b,
      /*c_mod=*/(short)0, c, /*reuse_a=*/false, /*reuse_b=*/false);
}

__device__ __forceinline__ void atomAddF(float* p, float v) {
  unsafeAtomicAdd(p, v);  // global_atomic_add_f32
}

// 16B global -> LDS async copy (per-lane), tracked by ASYNCcnt.
// lds_off = byte offset inside the kernel's (single) __shared__ array (LDS
// base offset 0). IMPORTANT: lds_base is passed as an extra (unreferenced)
// operand so the shared array ESCAPES into the asm — together with the
// "memory" clobber this makes the compiler treat the async DMA as a real LDS
// writer; otherwise the subsequent ds_load fragment reads fold to undef.
__device__ __forceinline__ void stage16(float* lds_base, unsigned lds_off_bytes,
                                        const float* gsrc) {
#if USE_ASYNC_LDS
  asm volatile("global_load_async_to_lds_b128 %0, %1, off"
               :: "v"(lds_off_bytes), "v"(gsrc), "v"(lds_base)
               : "memory");
#else
  *(float4*)((char*)lds_base + lds_off_bytes) = *(const float4*)gsrc;
#endif
}

__device__ __forceinline__ void async_wait() {
#if USE_ASYNC_LDS
  asm volatile("s_wait_asynccnt 0x0" ::: "memory");
#endif
}

// ---------------- small prep kernels ----------------

// Weff0 = W1 - W2 + W3 - W4   (cheb_w[0] minus admissible zero-response)
__global__ void weff0_kernel(float* __restrict__ w0, const float* __restrict__ cheb) {
  int i = blockIdx.x * blockDim.x + threadIdx.x;
  if (i >= CC * CC) return;
  w0[i] = cheb[1 * CC * CC + i] - cheb[2 * CC * CC + i]
        + cheb[3 * CC * CC + i] - cheb[4 * CC * CC + i];
}

__global__ void deg_kernel(float* __restrict__ deg,
                           const long long* __restrict__ src,
                           const long long* __restrict__ dst) {
  int e = blockIdx.x * blockDim.x + threadIdx.x;
  if (e >= EE) return;
  long long s = src[e], d = dst[e];
  if (s != d) atomAddF(&deg[s], 1.0f);
}

__global__ void dinv_kernel(float* __restrict__ dinv, const float* __restrict__ deg) {
  int n = blockIdx.x * blockDim.x + threadIdx.x;
  if (n >= NN) return;
  float dg = deg[n];
  dinv[n] = (dg > 0.0f) ? rsqrtf(fmaxf(dg, 1e-12f)) : 0.0f;
}

__global__ void we_kernel(float* __restrict__ we,
                          const long long* __restrict__ src,
                          const long long* __restrict__ dst,
                          const float* __restrict__ dinv,
                          const long long* __restrict__ batch,
                          const float* __restrict__ lmax) {
  int e = blockIdx.x * blockDim.x + threadIdx.x;
  if (e >= EE) return;
  long long s = src[e], d = dst[e];
  float w = 0.0f;
  if (s != d) {
    float le = lmax[batch[s]];
    w = -(2.0f / le) * dinv[s] * dinv[d];
  }
  we[e] = w;
}

// next = a2 * diag(n) * v + bprev * prev   (elementwise, float4; in-place safe)
__global__ void lhat_init(float* __restrict__ next, const float* __restrict__ v,
                          const float* __restrict__ prev,
                          const float* __restrict__ lmax, float a2, float bprev) {
  int i = blockIdx.x * blockDim.x + threadIdx.x;        // over N*C/4
  if (i >= NN * CC / 4) return;
  int n = i >> 5;                                       // node = (i*4)/128
  float diag = 2.0f / lmax[n >> 9] - 1.0f;              // 512 nodes/graph
  float4 vv = ((const float4*)v)[i];
  float4 pp = ((const float4*)prev)[i];
  float4 o;
  o.x = a2 * diag * vv.x + bprev * pp.x;
  o.y = a2 * diag * vv.y + bprev * pp.y;
  o.z = a2 * diag * vv.z + bprev * pp.z;
  o.w = a2 * diag * vv.w + bprev * pp.w;
  ((float4*)next)[i] = o;
}

// next[dst] += factor * w_e * v[src]  (4 channels per thread)
__global__ void lhat_scatter(float* __restrict__ next, const float* __restrict__ v,
                             const float* __restrict__ we,
                             const long long* __restrict__ src,
                             const long long* __restrict__ dst, float factor) {
  unsigned tid = blockIdx.x * blockDim.x + threadIdx.x; // E*32 threads
  int e  = tid >> 5;
  int c4 = (tid & 31) << 2;
  float w = we[e] * factor;
  if (w == 0.0f) return;
  long long s = src[e], d = dst[e];
  float4 vv = *(const float4*)&v[(size_t)s * CC + c4];
  float* p = &next[(size_t)d * CC + c4];
  atomAddF(p + 0, w * vv.x);
  atomAddF(p + 1, w * vv.y);
  atomAddF(p + 2, w * vv.z);
  atomAddF(p + 3, w * vv.w);
}

// ---------------- WMMA GEMMs (fp32 16x16x4) ----------------
// wave fragment layout: half = lane>>4, lr = lane&15
//   A: a.x = A[m0+lr][k0+2*half], a.y = A[m0+lr][k0+2*half+1]
//   B: b.x = B[k0+2*half][n0+lr], b.y = B[k0+2*half+1][n0+lr]
//   C/D reg r: row m0 + r + 8*half, col n0 + lr

// out[M,128] (+)= A[M,128] @ W[128,128]^T ; A-tile async-staged in LDS
__global__ void gemm_cheb(float* __restrict__ out, const float* __restrict__ A,
                          const float* __restrict__ W, int accum) {
  __shared__ float As[16 * 132];                        // padded: conflict-free
  const int m0 = blockIdx.x * 16;
  const int t = threadIdx.x;
#pragma unroll
  for (int j = t; j < 512; j += 256) {                  // 16x128 = 512 float4
    int row = j >> 5, c4 = (j & 31) << 2;
    stage16(As, (unsigned)((row * 132 + c4) * 4),
            &A[(size_t)(m0 + row) * CC + c4]);
  }
  async_wait();
  __syncthreads();
  const int wave = t >> 5, lane = t & 31, half = lane >> 4, lr = lane & 15;
  const int col = wave * 16 + lr;
  v8f c = {};
#pragma unroll
  for (int k0 = 0; k0 < CC; k0 += 4) {
    int ka = k0 + 2 * half;
    v2f a = *(const v2f*)&As[lr * 132 + ka];
    v2f b = *(const v2f*)&W[(size_t)col * CC + ka];     // B[k][col] = W[col][k]
    c = wmma_f32(a, b, c);
  }
#pragma unroll
  for (int r = 0; r < 8; ++r) {
    size_t idx = (size_t)(m0 + r + 8 * half) * CC + col;
    if (accum) out[idx] += c[r]; else out[idx] = c[r];
  }
}

// h[b] = U[b]^T @ x[b]  (TN). U chunk async-staged as Uc[k][m] (64x32, pad 36);
// each wave computes a 32x16 output (2 accumulators) so B fragments feed 2 WMMAs.
__global__ void gemm_uTx(float* __restrict__ h, const float* __restrict__ U,
                         const float* __restrict__ x) {
  __shared__ float Uc[64 * 36];
  const int g  = blockIdx.x >> 4;                       // 16 row-blocks/graph
  const int m0 = (blockIdx.x & 15) * 32;
  const int t = threadIdx.x, wave = t >> 5, lane = t & 31;
  const int half = lane >> 4, lr = lane & 15;
  const int col = wave * 16 + lr;
  const float* Ug = U + (size_t)g * NPGc * NPGc;
  const float* xg = x + (size_t)g * NPGc * CC;
  v8f c0 = {}, c1 = {};
  for (int k0 = 0; k0 < NPGc; k0 += 64) {
    // stage 64 k-rows x 32 m-cols of U (contiguous 128B rows)
#pragma unroll
    for (int j = t; j < 512; j += 256) {
      int row = j >> 3, c4 = (j & 7) << 2;
      stage16(Uc, (unsigned)((row * 36 + c4) * 4),
              &Ug[(size_t)(k0 + row) * NPGc + m0 + c4]);
    }
    async_wait();
    __syncthreads();
#pragma unroll
    for (int kk = 0; kk < 64; kk += 4) {
      int ka = kk + 2 * half;
      v2f a0, a1, b;
      a0.x = Uc[ka * 36 + lr];        a0.y = Uc[(ka + 1) * 36 + lr];
      a1.x = Uc[ka * 36 + 16 + lr];   a1.y = Uc[(ka + 1) * 36 + 16 + lr];
      int kb = k0 + ka;
      b.x = xg[(size_t)kb * CC + col];
      b.y = xg[(size_t)(kb + 1) * CC + col];
      c0 = wmma_f32(a0, b, c0);
      c1 = wmma_f32(a1, b, c1);
    }
    __syncthreads();
  }
  float* hg = h + (size_t)g * NPGc * CC;
#pragma unroll
  for (int r = 0; r < 8; ++r) {
    hg[(size_t)(m0 + r + 8 * half) * CC + col]      = c0[r];
    hg[(size_t)(m0 + 16 + r + 8 * half) * CC + col] = c1[r];
  }
}

// hs *= gaussian-smearing spectral filter (per eigen-row)
__global__ void smear_kernel(float* __restrict__ h, const float* __restrict__ Lambda,
                             const float* __restrict__ sw) {
  const int row = blockIdx.x;                           // b*512 + e
  const int c = threadIdx.x;                            // 128 threads
  const float lam = Lambda[row];
  __shared__ float gg[NG];
  const float step  = 2.0f / 31.0f;
  const float coeff = -0.5f / (step * step);
  if (c < NG) {
    float off = c * step;
    float d = lam - off;
    gg[c] = expf(coeff * d * d) - expf(coeff * off * off);
  }
  __syncthreads();
  float acc = 0.0f;
#pragma unroll
  for (int q = 0; q < NG; ++q) acc += gg[q] * sw[c * NG + q];
  float xn = lam * 0.5f;
  float tuk = (xn <= 0.5f) ? 1.0f
            : fmaxf(cosf(3.14159265358979323846f * (xn - 0.5f)), 0.0f);
  h[(size_t)row * CC + c] *= acc * tuk;
}

// out[b] += U[b] @ hs[b]  (NN). U row-block async-staged as Uc[m][k] (32x64,
// pad 68); 2 accumulators per wave share each hs fragment.
__global__ void gemm_uhs(float* __restrict__ out, const float* __restrict__ U,
                         const float* __restrict__ hs) {
  __shared__ float Uc[32 * 68];
  const int g  = blockIdx.x >> 4;
  const int m0 = (blockIdx.x & 15) * 32;
  const int t = threadIdx.x, wave = t >> 5, lane = t & 31;
  const int half = lane >> 4, lr = lane & 15;
  const int col = wave * 16 + lr;
  const float* Ug = U  + (size_t)g * NPGc * NPGc;
  const float* hg = hs + (size_t)g * NPGc * CC;
  float*       og = out + (size_t)g * NPGc * CC;
  v8f c0 = {}, c1 = {};
  for (int k0 = 0; k0 < NPGc; k0 += 64) {
#pragma unroll
    for (int j = t; j < 512; j += 256) {                // 32x64 = 512 float4
      int row = j >> 4, c4 = (j & 15) << 2;
      stage16(Uc, (unsigned)((row * 68 + c4) * 4),
              &Ug[(size_t)(m0 + row) * NPGc + k0 + c4]);
    }
    async_wait();
    __syncthreads();
#pragma unroll
    for (int kk = 0; kk < 64; kk += 4) {
      int ka = kk + 2 * half;
      v2f a0 = *(const v2f*)&Uc[lr * 68 + ka];
      v2f a1 = *(const v2f*)&Uc[(16 + lr) * 68 + ka];
      v2f b;
      int kb = k0 + ka;
      b.x = hg[(size_t)kb * CC + col];
      b.y = hg[(size_t)(kb + 1) * CC + col];
      c0 = wmma_f32(a0, b, c0);
      c1 = wmma_f32(a1, b, c1);
    }
    __syncthreads();
  }
#pragma unroll
  for (int r = 0; r < 8; ++r) {
    og[(size_t)(m0 + r + 8 * half) * CC + col]      += c0[r];
    og[(size_t)(m0 + 16 + r + 8 * half) * CC + col] += c1[r];
  }
}

// ---------------- launcher ----------------
extern "C" void kernel_launch(void* const* d_in, const int* in_sizes, int n_in,
                              void* d_out, int out_size, void* d_ws, size_t ws_size,
                              hipStream_t stream) {
  const float*     x     = (const float*)d_in[0];
  const long long* ei    = (const long long*)d_in[1];
  const float*     U     = (const float*)d_in[2];
  const float*     Lam   = (const float*)d_in[3];
  const float*     lmax  = (const float*)d_in[4];
  const long long* batch = (const long long*)d_in[5];
  const float*     cheb  = (const float*)d_in[6];
  const float*     sw    = (const float*)d_in[7];
  float* out = (float*)d_out;

  const long long* src = ei;
  const long long* dst = ei + EE;

  // workspace layout (floats)
  float* ws    = (float*)d_ws;
  float* weff0 = ws;                  // 16384
  float* deg   = weff0 + 16384;       // 32768
  float* dinv  = deg   + NN;          // 32768
  float* we    = dinv  + NN;          // 524288
  float* TxA   = we    + EE;          // 4194304
  float* TxB   = TxA   + NN * CC;     // 4194304  (total ~36 MB)
  float* h     = TxA;                 // reuse after spatial branch

  // prep
  weff0_kernel<<<64, 256, 0, stream>>>(weff0, cheb);
  hipMemsetAsync(deg, 0, NN * sizeof(float), stream);
  deg_kernel<<<EE / 256, 256, 0, stream>>>(deg, src, dst);
  dinv_kernel<<<NN / 256, 256, 0, stream>>>(dinv, deg);
  we_kernel<<<EE / 256, 256, 0, stream>>>(we, src, dst, dinv, batch, lmax);

  const int initG = NN * CC / 4 / 256;   // 4096
  const int scatG = EE * 32 / 256;       // 65536
  const int gemmG = NN / 16;             // 2048
  const int ugemG = NBg * 16;            // 1024

  // Tx1 = Lhat(x)
  lhat_init   <<<initG, 256, 0, stream>>>(TxA, x, x, lmax, 1.0f, 0.0f);
  lhat_scatter<<<scatG, 256, 0, stream>>>(TxA, x, we, src, dst, 1.0f);
  // out = x@Weff0^T + Tx1@W1^T
  gemm_cheb<<<gemmG, 256, 0, stream>>>(out, x,   weff0,            0);
  gemm_cheb<<<gemmG, 256, 0, stream>>>(out, TxA, cheb + 1 * CC*CC, 1);
  // Tx2 = 2*Lhat(Tx1) - x
  lhat_init   <<<initG, 256, 0, stream>>>(TxB, TxA, x, lmax, 2.0f, -1.0f);
  lhat_scatter<<<scatG, 256, 0, stream>>>(TxB, TxA, we, src, dst, 2.0f);
  gemm_cheb<<<gemmG, 256, 0, stream>>>(out, TxB, cheb + 2 * CC*CC, 1);
  // Tx3 = 2*Lhat(Tx2) - Tx1   (in-place into TxA)
  lhat_init   <<<initG, 256, 0, stream>>>(TxA, TxB, TxA, lmax, 2.0f, -1.0f);
  lhat_scatter<<<scatG, 256, 0, stream>>>(TxA, TxB, we, src, dst, 2.0f);
  gemm_cheb<<<gemmG, 256, 0, stream>>>(out, TxA, cheb + 3 * CC*CC, 1);
  // Tx4 = 2*Lhat(Tx3) - Tx2   (in-place into TxB)
  lhat_init   <<<initG, 256, 0, stream>>>(TxB, TxA, TxB, lmax, 2.0f, -1.0f);
  lhat_scatter<<<scatG, 256, 0, stream>>>(TxB, TxA, we, src, dst, 2.0f);
  gemm_cheb<<<gemmG, 256, 0, stream>>>(out, TxB, cheb + 4 * CC*CC, 1);

  // spectral: h = U^T x ; hs = h*s ; out += U @ hs
  gemm_uTx<<<ugemG, 256, 0, stream>>>(h, U, x);
  smear_kernel<<<NN, CC, 0, stream>>>(h, Lam, sw);
  gemm_uhs<<<ugemG, 256, 0, stream>>>(out, U, h);
}